// TT_Tables_46694884442289
// MI455X (gfx1250) — compile-verified
//
#include <hip/hip_runtime.h>

// ---- TT config (fixed by the reference) ----
#define NT        2        // NUM_TABLES
#define BATCH     1024
#define P0        216
#define P12       46656    // 216*216
#define R0        128
#define Q1R1      512      // Q1*R1 = 4*128
#define C0_ROW    256      // Q0*R0
#define C1_ROW    65536    // R0*Q1*R1
#define C2_ROW    256      // R1*Q2

typedef __attribute__((ext_vector_type(2))) float v2f;
typedef __attribute__((ext_vector_type(8))) float v8f;

// One workgroup per (table, batch) element.
//   Stage 1 (VALU, memory-bound): stream the 256KB core1 row with coalesced
//     b64 loads; AB[2][512] accumulated in registers, scattered to LDS as
//     the reshaped AB2[8][128].
//   Stage 2 (WMMA): wave 0 computes out[8][2] = AB2[8][128] x C[128][2]
//     with 32 x V_WMMA_F32_16X16X4_F32. All LDS reads use clamped in-bounds
//     addresses (unconditional ds_load_b64) + v_cndmask zeroing, so no
//     exec-mask predication is generated.
__global__ __launch_bounds__(256) void tt_embed_fwd(
    const int*   __restrict__ lS_i,    // [NT, BATCH]
    const float* __restrict__ core0,   // [NT, 216, 256]
    const float* __restrict__ core1,   // [NT, 216, 65536]
    const float* __restrict__ core2,   // [NT, 216, 256]
    float*       __restrict__ out)     // [NT, BATCH, 16]
{
    __shared__ __align__(16) float sA[C0_ROW];     // A:   [2][128] row-major
    __shared__ __align__(16) float sCt[2 * 128];   // C^T: [2][128]  (sCt[c][k] = C[k][c])
    __shared__ __align__(16) float sAB[8 * 128];   // AB2: [8][128]

    const int tid = threadIdx.x;
    const int b   = blockIdx.x & (BATCH - 1);
    const int t   = blockIdx.x >> 10;

    // mixed-radix index decomposition
    const int idx = lS_i[t * BATCH + b];
    const int i0  = idx / P12;
    const int i1  = (idx / P0) % P0;
    const int i2  = idx % P0;

    const float* arow = core0 + (size_t)(t * P0 + i0) * C0_ROW;
    const float* brow = core1 + (size_t)(t * P0 + i1) * C1_ROW;
    const float* crow = core2 + (size_t)(t * P0 + i2) * C2_ROW;

    // stage A (as-is) and C (transposed) into LDS
    sA[tid] = arow[tid];
    {
        const int k = tid & 127;       // R1 index
        const int c = tid >> 7;        // Q2 index (0..1)
        sCt[c * 128 + k] = crow[k * 2 + c];
    }
    __syncthreads();

    // ---------------- Stage 1: AB[2][512] ----------------
    // thread owns columns {2*tid, 2*tid+1}; per k-row the block reads a
    // contiguous 2KB segment (fully coalesced b64 loads).
    float acc00 = 0.f, acc01 = 0.f, acc10 = 0.f, acc11 = 0.f;
    const float2* bcol = (const float2*)brow + tid;   // + k*(Q1R1/2) per row
    #pragma unroll 8
    for (int k = 0; k < R0; ++k) {
        float2 bv = bcol[k * (Q1R1 / 2)];
        float  a0 = sA[k];          // A[0][k]
        float  a1 = sA[R0 + k];     // A[1][k]
        acc00 = fmaf(a0, bv.x, acc00);
        acc01 = fmaf(a0, bv.y, acc01);
        acc10 = fmaf(a1, bv.x, acc10);
        acc11 = fmaf(a1, bv.y, acc11);
    }

    // scatter to reshaped AB2[qq][r1]:  qq = q0*4 + (col>>7), r1 = col&127
    {
        const int c0 = 2 * tid, c1 = 2 * tid + 1;
        sAB[((c0 >> 7)    ) * 128 + (c0 & 127)] = acc00;
        sAB[((c1 >> 7)    ) * 128 + (c1 & 127)] = acc01;
        sAB[((c0 >> 7) + 4) * 128 + (c0 & 127)] = acc10;
        sAB[((c1 >> 7) + 4) * 128 + (c1 & 127)] = acc11;
    }
    __syncthreads();

    // ---------------- Stage 2: WMMA f32 16x16x4, K=128 ----------------
    if (tid < 32) {
        const int lane = tid;
        const int m    = lane & 15;         // A-matrix row (M)
        const int hi   = lane >> 4;         // K phase: VGPR0 K={k0,k0+2}, VGPR1 K={k0+1,k0+3}
        const int col  = lane & 15;         // B/D column (N)
        const int mrow = (m & 7) * 128;     // clamped (always in-bounds) A row base
        const int crowb = (col & 1) * 128;  // clamped C^T row base
        const bool mval = (m < 8);
        const bool cval = (col < 2);

        v8f acc0 = {}, acc1 = {};           // two chains to break WMMA C->C serialization
        #pragma unroll
        for (int kb = 0; kb < 32; kb += 2) {
            {
                const int k0 = kb * 4 + 2 * hi;
                float2 a2 = *(const float2*)&sAB[mrow + k0];   // ds_load_b64
                float2 c2 = *(const float2*)&sCt[crowb + k0];  // ds_load_b64
                v2f av = { mval ? a2.x : 0.0f, mval ? a2.y : 0.0f };
                v2f bv = { cval ? c2.x : 0.0f, cval ? c2.y : 0.0f };
                acc0 = __builtin_amdgcn_wmma_f32_16x16x4_f32(
                    false, av, false, bv, (short)0, acc0, false, false);
            }
            {
                const int k0 = (kb + 1) * 4 + 2 * hi;
                float2 a2 = *(const float2*)&sAB[mrow + k0];
                float2 c2 = *(const float2*)&sCt[crowb + k0];
                v2f av = { mval ? a2.x : 0.0f, mval ? a2.y : 0.0f };
                v2f bv = { cval ? c2.x : 0.0f, cval ? c2.y : 0.0f };
                acc1 = __builtin_amdgcn_wmma_f32_16x16x4_f32(
                    false, av, false, bv, (short)0, acc1, false, false);
            }
        }

        // D layout: lanes 0..15, VGPR j -> (M=j, N=lane). Useful: M<8, N<2.
        if (hi == 0 && col < 2) {
            float* op = out + ((size_t)t * BATCH + b) * 16;
            #pragma unroll
            for (int j = 0; j < 8; ++j)
                op[j * 2 + col] = acc0[j] + acc1[j];   // out[b][qq*2 + q2]
        }
    }
}

extern "C" void kernel_launch(void* const* d_in, const int* in_sizes, int n_in,
                              void* d_out, int out_size, void* d_ws, size_t ws_size,
                              hipStream_t stream) {
    const int*   lS_i  = (const int*)d_in[0];     // [2,1024] (integer per harness convention)
    const float* core0 = (const float*)d_in[1];   // [2,216,256]
    const float* core1 = (const float*)d_in[2];   // [2,216,65536]
    const float* core2 = (const float*)d_in[3];   // [2,216,256]
    float* out = (float*)d_out;                   // [2,1024,16]

    (void)in_sizes; (void)n_in; (void)d_ws; (void)ws_size; (void)out_size;

    tt_embed_fwd<<<dim3(NT * BATCH), dim3(256), 0, stream>>>(
        lS_i, core0, core1, core2, out);
}